// SinkhornLoss_69148973465955
// MI455X (gfx1250) — compile-verified
//
#include <hip/hip_runtime.h>
#include <math.h>

// Sinkhorn loss, m=8, n=2048, d=3 (fp32).
// M (8x2048x2048, 134 MB) is never materialized: coordinates (+|.|^2) live in
// LDS, 16x16 tiles of the Gram matrix come from V_WMMA_F32_16X16X4_F32
// (d=3 padded to K=4, full f32 precision -- required since eps=1e-3 amplifies
// M errors by 1000x in the exponent), and the distance + online-LSE math is
// fused on the WMMA accumulators. All Sinkhorn state stays LDS-resident, so
// the 50-iteration loop does zero HBM traffic; the kernel is exp/VALU bound.
// Inner loop uses a single v_exp_f32 per element (one of the two LSE rescale
// exponents is always exactly 0) and ds_load_b128 for the B-tile fetch.

#define N_PTS 2048
#define DIM 3
#define EPSILON 1e-3f
#define INV_EPS (1.0f / EPSILON)
#define MAX_ITERS 50
#define STOP_THRESH 1e-3f
#define BLOCK 1024
#define NWAVES (BLOCK / 32)
#define NTILES (N_PTS / 16) // 128 tiles of 16

typedef __attribute__((ext_vector_type(2))) float v2f;
typedef __attribute__((ext_vector_type(8))) float v8f;

// Online LSE update with exactly one exp:
//   nm = max(mx,t); e = exp(min-max); sm' = t>mx ? sm*e+1 : sm+e
__device__ __forceinline__ void lse_update(float& mx, float& sm, float t) {
  const float nm = fmaxf(mx, t);
  const float e  = __expf(fminf(mx, t) - nm);
  sm = (t > mx) ? fmaf(sm, e, 1.0f) : (sm + e);
  mx = nm;
}

// Merge two (mx,sm) LSE states with one exp.
__device__ __forceinline__ void lse_merge(float& mx, float& sm, float m2, float s2) {
  const float nm = fmaxf(mx, m2);
  const float e  = __expf(fminf(mx, m2) - nm);
  sm = (m2 > mx) ? fmaf(sm, e, s2) : fmaf(s2, e, sm);
  mx = nm;
}

// One Sinkhorn half-iteration:
//   out[i] = EPS * (logw[i] - LSE_j((dualQ[j] - D(i,j)) / EPS))
// with D(i,j) = max(|P_i|^2 + |Q_j|^2 - 2 P_i.Q_j, 0). All LSE math is done
// in t-units (pre-scaled by 1/EPS; valid since 1/EPS > 0 commutes with max).
// P,Q: LDS [N_PTS][4] floats, slots 0..2 = coords, slot 3 = |.|^2.
__device__ __forceinline__ void lse_pass(
    const float* P, const float* Q,
    const float* dualQ, const float* logw, float* out,
    int wave, int lane)
{
  const int lhalf = lane >> 4;  // 0: K={0,1} half, 1: K={2,3} half
  const int l15   = lane & 15;
  const float4* Q4 = (const float4*)Q;

  for (int rt = wave; rt < NTILES; rt += NWAVES) {
    const int j0 = rt * 16;

    // A tile (16x4 rows of P): lanes 0-15 hold K=0,1; lanes 16-31 hold K=2,3.
    const float4 pa = ((const float4*)P)[j0 + l15];
    v2f A;
    A.x = lhalf ? pa.z : pa.x;
    A.y = lhalf ? 0.0f : pa.y;

    float p2s[8], mx[8], sm[8];
#pragma unroll
    for (int r = 0; r < 8; ++r) {
      p2s[r] = P[(j0 + lhalf * 8 + r) * 4 + 3] * INV_EPS; // |P_row|^2 / eps
      mx[r]  = -__builtin_inff();
      sm[r]  = 0.0f;
    }

#pragma unroll 2
    for (int kt = 0; kt < NTILES; ++kt) {
      const int col = kt * 16 + l15;
      const float4 q = Q4[col];           // one ds_load_b128: coords + norm
      v2f B;
      B.x = lhalf ? q.z : q.x;
      B.y = lhalf ? 0.0f : q.y;
      const float q2s = q.w * INV_EPS;
      const float dvs = dualQ[col] * INV_EPS;

      v8f C = {};
      C = __builtin_amdgcn_wmma_f32_16x16x4_f32(
          false, A, false, B, (short)0, C, false, false);

#pragma unroll
      for (int r = 0; r < 8; ++r) {
        // tM = D(i,j)/eps ; t = (dual - D)/eps
        const float tM = fmaxf(fmaf(-2.0f * INV_EPS, C[r], p2s[r] + q2s), 0.0f);
        lse_update(mx[r], sm[r], dvs - tM);
      }
    }

    // Merge LSE state across the 16 lanes holding each row (within halves).
#pragma unroll
    for (int mask = 1; mask <= 8; mask <<= 1) {
#pragma unroll
      for (int r = 0; r < 8; ++r) {
        const float m2 = __shfl_xor(mx[r], mask, 32);
        const float s2 = __shfl_xor(sm[r], mask, 32);
        lse_merge(mx[r], sm[r], m2, s2);
      }
    }

    if (l15 == 0) {
#pragma unroll
      for (int r = 0; r < 8; ++r) {
        const int j = j0 + lhalf * 8 + r;
        const float lse = mx[r] + __logf(sm[r]);
        out[j] = EPSILON * (logw[j] - lse);
      }
    }
  }
}

// loss partial: sum over this wave's tiles of M * exp((u+v-M)/eps).
// Computed in t-units: EPS * sum( tM * exp(us + vs - tM) ), tM = M/eps.
__device__ __forceinline__ float loss_pass(
    const float* P, const float* Q,
    const float* u, const float* v, int wave, int lane)
{
  const int lhalf = lane >> 4;
  const int l15   = lane & 15;
  const float4* Q4 = (const float4*)Q;
  float acc = 0.0f;

  for (int rt = wave; rt < NTILES; rt += NWAVES) {
    const int j0 = rt * 16;
    const float4 pa = ((const float4*)P)[j0 + l15];
    v2f A;
    A.x = lhalf ? pa.z : pa.x;
    A.y = lhalf ? 0.0f : pa.y;

    float p2s[8], us[8];
#pragma unroll
    for (int r = 0; r < 8; ++r) {
      const int j = j0 + lhalf * 8 + r;
      p2s[r] = P[j * 4 + 3] * INV_EPS;
      us[r]  = u[j] * INV_EPS;
    }

#pragma unroll 2
    for (int kt = 0; kt < NTILES; ++kt) {
      const int col = kt * 16 + l15;
      const float4 q = Q4[col];
      v2f B;
      B.x = lhalf ? q.z : q.x;
      B.y = lhalf ? 0.0f : q.y;
      const float q2s = q.w * INV_EPS;
      const float vks = v[col] * INV_EPS;

      v8f C = {};
      C = __builtin_amdgcn_wmma_f32_16x16x4_f32(
          false, A, false, B, (short)0, C, false, false);

#pragma unroll
      for (int r = 0; r < 8; ++r) {
        const float tM = fmaxf(fmaf(-2.0f * INV_EPS, C[r], p2s[r] + q2s), 0.0f);
        acc = fmaf(tM, __expf(us[r] + vks - tM), acc);
      }
    }
  }
  return acc * EPSILON;
}

__global__ void __launch_bounds__(BLOCK)
sinkhorn_loss_kernel(const float* __restrict__ pred,
                     const float* __restrict__ expd,
                     const float* __restrict__ aw,
                     const float* __restrict__ bw,
                     float* __restrict__ out)
{
  extern __shared__ float smem[];
  float* xp = smem;                 // [N_PTS][4]: x0,x1,x2,|x|^2
  float* yp = xp + N_PTS * 4;       // [N_PTS][4]
  float* la = yp + N_PTS * 4;       // log a
  float* lb = la + N_PTS;           // log b
  float* u  = lb + N_PTS;
  float* v  = u  + N_PTS;
  float* un = v  + N_PTS;
  float* vn = un + N_PTS;
  float* red = vn + N_PTS;          // [NWAVES]
  int*   doneF = (int*)(red + NWAVES);

  const int m   = blockIdx.x;
  const int tid = threadIdx.x;
  const int wave = tid >> 5;
  const int lane = tid & 31;

  const float* X = pred + (size_t)m * N_PTS * DIM;
  const float* Y = expd + (size_t)m * N_PTS * DIM;

  for (int i = tid; i < N_PTS; i += BLOCK) {
    const float x0 = X[i * 3 + 0], x1 = X[i * 3 + 1], x2c = X[i * 3 + 2];
    xp[i * 4 + 0] = x0; xp[i * 4 + 1] = x1; xp[i * 4 + 2] = x2c;
    xp[i * 4 + 3] = fmaf(x0, x0, fmaf(x1, x1, x2c * x2c));
    const float y0 = Y[i * 3 + 0], y1 = Y[i * 3 + 1], y2c = Y[i * 3 + 2];
    yp[i * 4 + 0] = y0; yp[i * 4 + 1] = y1; yp[i * 4 + 2] = y2c;
    yp[i * 4 + 3] = fmaf(y0, y0, fmaf(y1, y1, y2c * y2c));
    la[i] = __logf(aw[(size_t)m * N_PTS + i]);
    lb[i] = __logf(bw[(size_t)m * N_PTS + i]);
    u[i] = 0.0f; v[i] = 0.0f; un[i] = 0.0f; vn[i] = 0.0f;
  }
  if (tid == 0) *doneF = 0;
  __syncthreads();

  for (int it = 0; it < MAX_ITERS; ++it) {
    if (*doneF) break; // uniform: written before the trailing barrier

    // u_new = eps*(log_a - LSE_k((v - M)/eps))
    lse_pass(xp, yp, v, la, un, wave, lane);
    __syncthreads();
    // v_new = eps*(log_b - LSE_j((u_new - M)/eps)); roles of x,y swap
    lse_pass(yp, xp, un, lb, vn, wave, lane);
    __syncthreads();

    // err = max(|u_new - u|, |v_new - v|); then u,v <- new
    float e = 0.0f;
    for (int i = tid; i < N_PTS; i += BLOCK) {
      e = fmaxf(e, fabsf(un[i] - u[i]));
      e = fmaxf(e, fabsf(vn[i] - v[i]));
      u[i] = un[i];
      v[i] = vn[i];
    }
#pragma unroll
    for (int mask = 1; mask <= 16; mask <<= 1)
      e = fmaxf(e, __shfl_xor(e, mask, 32));
    if (lane == 0) red[wave] = e;
    __syncthreads();
    if (tid == 0) {
      float me = 0.0f;
      for (int w = 0; w < NWAVES; ++w) me = fmaxf(me, red[w]);
      *doneF = (me < STOP_THRESH) ? 1 : 0;
    }
    __syncthreads();
  }

  // loss[m] = sum_{j,k} M * exp((-M + u + v)/eps)
  float acc = loss_pass(xp, yp, u, v, wave, lane);
#pragma unroll
  for (int mask = 1; mask <= 16; mask <<= 1)
    acc += __shfl_xor(acc, mask, 32);
  if (lane == 0) red[wave] = acc;
  __syncthreads();
  if (tid == 0) {
    float s = 0.0f;
    for (int w = 0; w < NWAVES; ++w) s += red[w];
    out[m] = s;
  }
}

extern "C" void kernel_launch(void* const* d_in, const int* in_sizes, int n_in,
                              void* d_out, int out_size, void* d_ws, size_t ws_size,
                              hipStream_t stream) {
  (void)n_in; (void)d_ws; (void)ws_size; (void)out_size;
  const float* pred = (const float*)d_in[0];
  const float* expd = (const float*)d_in[1];
  const float* aw   = (const float*)d_in[2];
  const float* bw   = (const float*)d_in[3];
  float* out = (float*)d_out;

  const int m = in_sizes[0] / (N_PTS * DIM); // 8

  // LDS: 2*[2048][4] coords+norm + 6*[2048] vectors + reduction scratch
  const size_t shmem =
      (size_t)(N_PTS * 4 * 2 + N_PTS * 6 + NWAVES) * sizeof(float) + 256;

  (void)hipFuncSetAttribute((const void*)sinkhorn_loss_kernel,
                            hipFuncAttributeMaxDynamicSharedMemorySize,
                            (int)shmem);

  sinkhorn_loss_kernel<<<dim3(m), dim3(BLOCK), shmem, stream>>>(
      pred, expd, aw, bw, out);
}